// CenterLoss_64699387347720
// MI455X (gfx1250) — compile-verified
//
#include <hip/hip_runtime.h>
#include <hip/hip_bf16.h>
#include <stdint.h>

#define NUM_CLASSES 6625
#define FEAT_DIM 96
#define NROWS 4096
#define CLAMP_MIN 1e-12
#define CLAMP_MAX 1e12

typedef float v2f __attribute__((ext_vector_type(2)));
typedef float v4f __attribute__((ext_vector_type(4)));
typedef float v8f __attribute__((ext_vector_type(8)));

// ---------------------------------------------------------------------------
// Kernel 1: per-row argmax over predicts [NROWS x NUM_CLASSES], first-max tie
// rule (faithful to jnp.argmax). One 256-thread block (8 wave32) per row.
// Pure streaming: scalar prologue to 16B alignment, b128 NT loads, scalar tail.
// ---------------------------------------------------------------------------
__global__ __launch_bounds__(256) void argmax_rows_kernel(
    const float* __restrict__ predicts, int* __restrict__ labels) {
  const int row = blockIdx.x;
  const int tid = threadIdx.x;
  const float* p = predicts + (size_t)row * NUM_CLASSES;

  float best = -__builtin_inff();
  int bidx = NUM_CLASSES;  // sentinel larger than any real index

  // scalar prologue to 16-byte alignment (row base is only 4B aligned)
  const int pro = (int)(((16u - ((uint32_t)(uintptr_t)p & 15u)) & 15u) >> 2);
  if (tid < pro) {
    float x = p[tid];
    if (x > best) { best = x; bidx = tid; }
  }

  // vector body: b128 non-temporal loads (data read exactly once)
  const float* pa = p + pro;
  const int nrem = NUM_CLASSES - pro;
  const int nvec = nrem >> 2;
  for (int v = tid; v < nvec; v += 256) {
    v4f x = __builtin_nontemporal_load((const v4f*)pa + v);
    const int base = pro + (v << 2);
    if (x.x > best) { best = x.x; bidx = base; }
    if (x.y > best) { best = x.y; bidx = base + 1; }
    if (x.z > best) { best = x.z; bidx = base + 2; }
    if (x.w > best) { best = x.w; bidx = base + 3; }
  }

  // scalar tail
  {
    const int idx = pro + (nvec << 2) + tid;
    if (idx < NUM_CLASSES) {
      float x = p[idx];
      if (x > best || (x == best && idx < bidx)) { best = x; bidx = idx; }
    }
  }

  // wave32 reduction, prefer smaller index on ties
  #pragma unroll
  for (int off = 16; off; off >>= 1) {
    float ov = __shfl_xor(best, off, 32);
    int   oi = __shfl_xor(bidx, off, 32);
    if (ov > best || (ov == best && oi < bidx)) { best = ov; bidx = oi; }
  }

  __shared__ float sv[8];
  __shared__ int   si[8];
  if ((tid & 31) == 0) { sv[tid >> 5] = best; si[tid >> 5] = bidx; }
  __syncthreads();
  if (tid == 0) {
    best = sv[0]; bidx = si[0];
    #pragma unroll
    for (int w = 1; w < 8; ++w) {
      if (sv[w] > best || (sv[w] == best && si[w] < bidx)) {
        best = sv[w]; bidx = si[w];
      }
    }
    labels[row] = bidx;
  }
}

// ---------------------------------------------------------------------------
// Kernel 2: per-sample squared distance to its labeled center.
// One wave (32 lanes) handles a tile of 16 samples. The cross term
// diag(feats_tile @ gathered_centers^T) is computed with chained
// V_WMMA_F32_16X16X4_F32 over K = 96 (24 steps). Squared norms are
// accumulated per-lane in f64 and combined across the two lane-halves.
//
// A (16x4 f32, ISA 7.12.2): lanes 0-15 hold M=lane, V0=K0 / V1=K1;
//                           lanes 16-31 hold M=lane-16, V0=K2 / V1=K3.
// B (4x16):                 lanes 0-15 hold N=lane, V0=K0 / V1=K1;
//                           lanes 16-31 hold N=lane-16, V0=K2 / V1=K3.
// C/D (16x16 f32): lanes 0-15: VGPR r -> M=r; lanes 16-31: VGPR r -> M=8+r.
// ---------------------------------------------------------------------------
__global__ __launch_bounds__(32) void center_dist_kernel(
    const float* __restrict__ feats, const float* __restrict__ centers,
    const int* __restrict__ labels, double* __restrict__ dist) {
  __shared__ float cmat[16 * 17];  // padded to dodge bank conflicts

  const int lane = threadIdx.x;
  const int m0 = blockIdx.x * 16;
  const int row = lane & 15;            // sample within tile (M for A, N for B)
  const int colOff = (lane >> 4) * 2;   // lanes 0-15 -> K{0,1}, lanes 16-31 -> K{2,3}

  const int lab = labels[m0 + row];
  const float* fp = feats + (size_t)(m0 + row) * FEAT_DIM + colOff;
  const float* cp = centers + (size_t)lab * FEAT_DIM + colOff;

  v8f acc = {};
  double sqf = 0.0, sqc = 0.0;

  #pragma unroll
  for (int t = 0; t < FEAT_DIM / 4; ++t) {
    v2f a = *(const v2f*)(fp + 4 * t);  // 8B aligned: 4t+colOff is even
    v2f b = *(const v2f*)(cp + 4 * t);
    sqf += (double)a.x * a.x + (double)a.y * a.y;
    sqc += (double)b.x * b.x + (double)b.y * b.y;
    // D = A x B + C, f32 16x16x4
    acc = __builtin_amdgcn_wmma_f32_16x16x4_f32(
        /*neg_a=*/false, a, /*neg_b=*/false, b,
        /*c_mod=*/(short)0, acc, /*reuse_a=*/false, /*reuse_b=*/false);
  }

  // each lane saw half the K columns of its sample; fold halves (wave32)
  sqf += __shfl_xor(sqf, 16, 32);
  sqc += __shfl_xor(sqc, 16, 32);

  // spill C to LDS, pull the diagonal
  const int mBase = (lane >> 4) * 8;
  const int n = lane & 15;
  #pragma unroll
  for (int r = 0; r < 8; ++r) cmat[(mBase + r) * 17 + n] = acc[r];
  __syncthreads();

  if (lane < 16) {
    const double dot = (double)cmat[lane * 17 + lane];
    double d = sqf + sqc - 2.0 * dot;
    d = d < CLAMP_MIN ? CLAMP_MIN : (d > CLAMP_MAX ? CLAMP_MAX : d);
    dist[m0 + lane] = d;
  }
}

// ---------------------------------------------------------------------------
// Kernel 3: reduce 4096 doubles, normalize, add the constant contribution of
// the (C-1) masked-to-zero-then-clamped columns per sample.
// ---------------------------------------------------------------------------
__global__ __launch_bounds__(256) void finalize_kernel(
    const double* __restrict__ dist, float* __restrict__ out) {
  __shared__ double sm[256];
  double s = 0.0;
  for (int i = threadIdx.x; i < NROWS; i += 256) s += dist[i];
  sm[threadIdx.x] = s;
  __syncthreads();
  #pragma unroll
  for (int stride = 128; stride > 0; stride >>= 1) {
    if (threadIdx.x < stride) sm[threadIdx.x] += sm[threadIdx.x + stride];
    __syncthreads();
  }
  if (threadIdx.x == 0) {
    const double loss =
        sm[0] / (double)NROWS + (double)(NUM_CLASSES - 1) * CLAMP_MIN;
    out[0] = (float)loss;
  }
}

extern "C" void kernel_launch(void* const* d_in, const int* in_sizes, int n_in,
                              void* d_out, int out_size, void* d_ws, size_t ws_size,
                              hipStream_t stream) {
  const float* feats    = (const float*)d_in[0];  // [64,64,96]
  const float* predicts = (const float*)d_in[1];  // [64,64,6625]
  const float* centers  = (const float*)d_in[2];  // [6625,96]

  int*    labels = (int*)d_ws;                                   // 4096 ints
  double* dist   = (double*)((char*)d_ws + NROWS * sizeof(int)); // 4096 doubles (8B aligned)

  argmax_rows_kernel<<<NROWS, 256, 0, stream>>>(predicts, labels);
  center_dist_kernel<<<NROWS / 16, 32, 0, stream>>>(feats, centers, labels, dist);
  finalize_kernel<<<1, 256, 0, stream>>>(dist, (float*)d_out);
}